// CompressionRelationPredictor_35536559407708
// MI455X (gfx1250) — compile-verified
//
#include <hip/hip_runtime.h>

// ---------------------------------------------------------------------------
// RGCN forward on MI455X (gfx1250, wave32).
// All hot data (x tiles 12.8MB, W 0.7MB, out 12.8MB) is L2-resident (192MB),
// so the layer scatter is L2-atomic-bound; matrix math uses fp32 WMMA
// (v_wmma_f32_16x16x4_f32) to keep reference fp32 numerics. A-tile staging in
// the message kernel uses the gfx1250 async VMEM->LDS path when available.
// ---------------------------------------------------------------------------

#define N_NODES 50000
#define NRELS   20
#define RTOT    41            // 2*NRELS + 1
#define NEMB    128
#define NHID    64
#define NEDGE   1600000
#define NTRIP   200000
#define EAUG    (2*NEDGE + N_NODES)   // 3,250,000 augmented edges
#define NGROUPS ((EAUG + 15) / 16)    // 203,125 groups of 16 edges
#define WPB     4                     // waves per block in message kernel

#if defined(__has_builtin)
#  if __has_builtin(__builtin_amdgcn_global_load_async_to_lds_b128) && \
      __has_builtin(__builtin_amdgcn_s_wait_asynccnt)
#    define HAVE_ASYNC_LDS 1
#  endif
#endif
#ifndef HAVE_ASYNC_LDS
#  define HAVE_ASYNC_LDS 0
#endif

typedef __attribute__((ext_vector_type(2))) float v2f;
typedef __attribute__((ext_vector_type(8))) float v8f;

#if HAVE_ASYNC_LDS
// builtin expects: (int __vector(4) __device__*, int __vector(4) __shared__*, imm, imm)
typedef int v4i_vs __attribute__((vector_size(16)));
typedef v4i_vs __attribute__((address_space(1)))* async_gptr_t;
typedef v4i_vs __attribute__((address_space(3)))* async_lptr_t;
#endif

// augmented edge e -> (src, rel, tgt), matching reference _augment()
__device__ __forceinline__ void edge_at(const int* __restrict__ g, int e,
                                        int& s, int& r, int& t) {
  if (e < NEDGE)            { s = g[3*e];       r = g[3*e+1];          t = g[3*e+2]; }
  else if (e < 2*NEDGE)     { int i = e-NEDGE; s = g[3*i+2]; r = g[3*i+1]+NRELS; t = g[3*i]; }
  else                      { int n = e-2*NEDGE; s = n; r = 2*NRELS; t = n; }
}

__global__ void k_zero(float* __restrict__ p, long long n) {
  long long i = (long long)blockIdx.x * blockDim.x + threadIdx.x;
  if (i < n) p[i] = 0.0f;
}

// degree counts (float atomics; max degree < 2^24 so exact)
__global__ void k_degree(const int* __restrict__ g, float* __restrict__ deg1,
                         float* __restrict__ deg2, int* __restrict__ relCount) {
  int e = blockIdx.x * blockDim.x + threadIdx.x;
  if (e >= EAUG) return;
  int s, r, t; edge_at(g, e, s, r, t);
  (void)s;
  atomicAdd(&deg1[t], 1.0f);
  atomicAdd(&deg2[(size_t)r * N_NODES + t], 1.0f);
  atomicAdd(&relCount[r], 1);
}

__global__ void k_scan(const int* __restrict__ relCount, int* __restrict__ relOffset,
                       int* __restrict__ relCursor) {
  if (blockIdx.x == 0 && threadIdx.x == 0) {
    int acc = 0;
    for (int r = 0; r < RTOT; ++r) {
      relOffset[r] = acc; relCursor[r] = acc; acc += relCount[r];
    }
    relOffset[RTOT] = acc;
  }
}

__global__ void k_bucket(const int* __restrict__ g, int* __restrict__ relCursor,
                         int* __restrict__ perm) {
  int e = blockIdx.x * blockDim.x + threadIdx.x;
  if (e >= EAUG) return;
  int s, r, t; edge_at(g, e, s, r, t);
  (void)s; (void)t;
  int pos = atomicAdd(&relCursor[r], 1);
  perm[pos] = e;
}

// x0 = emb(N x 128) @ We(128 x 64) + be ; one wave per 16-row tile (N = 3125*16)
__global__ void __launch_bounds__(32)
k_encode(const float* __restrict__ emb, const float* __restrict__ We,
         const float* __restrict__ be, float* __restrict__ x0) {
  int tile  = blockIdx.x;
  int lane  = threadIdx.x;
  int m     = lane & 15;
  int khalf = (lane >> 4) * 2;           // lanes 16-31 hold K+2,K+3
  v8f z = {};
  v8f acc0 = z, acc1 = z, acc2 = z, acc3 = z;
  const float* arow = emb + (size_t)(tile * 16 + m) * NEMB;
  for (int k = 0; k < NEMB; k += 4) {
    int kk = k + khalf;
    v2f a; a.x = arow[kk]; a.y = arow[kk + 1];
    const float* brow0 = We + (size_t)kk * NHID;
    const float* brow1 = We + (size_t)(kk + 1) * NHID;
    v2f b;
    b.x = brow0[m +  0]; b.y = brow1[m +  0];
    acc0 = __builtin_amdgcn_wmma_f32_16x16x4_f32(false, a, false, b, (short)0, acc0, false, false);
    b.x = brow0[m + 16]; b.y = brow1[m + 16];
    acc1 = __builtin_amdgcn_wmma_f32_16x16x4_f32(false, a, false, b, (short)0, acc1, false, false);
    b.x = brow0[m + 32]; b.y = brow1[m + 32];
    acc2 = __builtin_amdgcn_wmma_f32_16x16x4_f32(false, a, false, b, (short)0, acc2, false, false);
    b.x = brow0[m + 48]; b.y = brow1[m + 48];
    acc3 = __builtin_amdgcn_wmma_f32_16x16x4_f32(false, a, false, b, (short)0, acc3, false, false);
  }
  int rowSel = (lane < 16) ? 0 : 8;      // C/D layout: VGPR j = row j / j+8
#pragma unroll
  for (int j = 0; j < 8; ++j) {
    size_t row = (size_t)(tile * 16 + rowSel + j) * NHID;
    x0[row + m +  0] = acc0[j] + be[m +  0];
    x0[row + m + 16] = acc1[j] + be[m + 16];
    x0[row + m + 32] = acc2[j] + be[m + 32];
    x0[row + m + 48] = acc3[j] + be[m + 48];
  }
}

// One wave per group of 16 relation-bucketed edges:
//   gather 16 x[src] rows -> LDS (async VMEM->LDS when available),
//   16x64x64 WMMA GEMM with W[rel], scale by degree norm,
//   atomic scatter into out[tgt] (atomics resolve in L2).
__global__ void __launch_bounds__(32 * WPB)
k_message(const float* __restrict__ x, const float* __restrict__ Wall,
          const float* __restrict__ deg1, const float* __restrict__ deg2,
          const int* __restrict__ g, const int* __restrict__ perm,
          float* __restrict__ out, int vertical) {
  __shared__ __align__(16) float sX[WPB][16][NHID];
  __shared__ int   sTgt[WPB][16];
  __shared__ int   sRel[WPB][16];
  __shared__ int   sSrc[WPB][16];
  __shared__ float sW[WPB][16];

  int wv   = threadIdx.x >> 5;
  int lane = threadIdx.x & 31;
  int grp  = blockIdx.x * WPB + wv;
  int base = grp * 16;
  int count = 0;
  if (grp < NGROUPS) { count = EAUG - base; if (count > 16) count = 16; }

  if (lane < 16) {
    int s = 0, r = 0, t = 0; float w = 0.0f;
    if (lane < count) {
      int e = perm[base + lane];
      edge_at(g, e, s, r, t);
      float d = vertical ? deg2[(size_t)r * N_NODES + t] : deg1[t];
      w = 1.0f / fmaxf(d, 1.0f);
    }
    sSrc[wv][lane] = s; sTgt[wv][lane] = t; sRel[wv][lane] = r; sW[wv][lane] = w;
  }
  __syncthreads();

  // stage A tile (16 gathered rows of x, 4KB) into this wave's LDS region
#if HAVE_ASYNC_LDS
  // gfx1250 async VMEM->LDS: 8 issues x 32 lanes x 16B, no VGPR round-trip,
  // tracked by ASYNCcnt (wave-local staging => wave-local wait suffices)
#pragma unroll
  for (int it = 0; it < 8; ++it) {
    int idx  = it * 32 + lane;          // [0,256) float4 elements
    int row  = idx >> 4;
    int col4 = idx & 15;
    if (row < count) {
      const float* gsrc = x + (size_t)sSrc[wv][row] * NHID + col4 * 4;
      __builtin_amdgcn_global_load_async_to_lds_b128(
          (async_gptr_t)(void*)const_cast<float*>(gsrc),
          (async_lptr_t)(void*)&sX[wv][row][col4 * 4],
          0, 0);
    } else {
      float4 z4 = {0.f, 0.f, 0.f, 0.f};
      *(float4*)&sX[wv][row][col4 * 4] = z4;
    }
  }
  __builtin_amdgcn_s_wait_asynccnt(0);
#else
  for (int idx = lane; idx < 16 * (NHID / 4); idx += 32) {
    int row  = idx >> 4;
    int col4 = idx & 15;
    float4 v = {0.f, 0.f, 0.f, 0.f};
    if (row < count) v = ((const float4*)(x + (size_t)sSrc[wv][row] * NHID))[col4];
    *(float4*)&sX[wv][row][col4 * 4] = v;
  }
#endif
  bool uniform = true;
  int r0 = sRel[wv][0];
  for (int i = 1; i < count; ++i) uniform = uniform && (sRel[wv][i] == r0);
  __syncthreads();

  if (count == 0) return;

  int m     = lane & 15;
  int khalf = (lane >> 4) * 2;
  if (uniform) {
    const float* W = Wall + (size_t)r0 * NHID * NHID;
    v8f z = {};
    v8f acc[4] = {z, z, z, z};
    for (int k = 0; k < NHID; k += 4) {
      int kk = k + khalf;
      v2f a = *(const v2f*)&sX[wv][m][kk];      // one ds_load_b64 per k-step
      const float* brow0 = W + (size_t)kk * NHID;
      const float* brow1 = W + (size_t)(kk + 1) * NHID;
#pragma unroll
      for (int nb = 0; nb < 4; ++nb) {
        v2f b; b.x = brow0[nb * 16 + m]; b.y = brow1[nb * 16 + m];
        acc[nb] = __builtin_amdgcn_wmma_f32_16x16x4_f32(false, a, false, b, (short)0, acc[nb], false, false);
      }
    }
    int rowSel = (lane < 16) ? 0 : 8;
#pragma unroll
    for (int nb = 0; nb < 4; ++nb)
#pragma unroll
      for (int j = 0; j < 8; ++j) {
        int row = rowSel + j;
        if (row < count) {
          int col = nb * 16 + m;
          atomicAdd(&out[(size_t)sTgt[wv][row] * NHID + col], sW[wv][row] * acc[nb][j]);
        }
      }
  } else {
    // rare bucket-boundary group (<= RTOT-1 of these): scalar matvec per edge
    for (int i = 0; i < count; ++i) {
      const float* W = Wall + (size_t)sRel[wv][i] * NHID * NHID;
      float wgt = sW[wv][i];
      int   t   = sTgt[wv][i];
      for (int c = lane; c < NHID; c += 32) {
        float a = 0.0f;
        for (int h = 0; h < NHID; ++h) a += sX[wv][i][h] * W[(size_t)h * NHID + c];
        atomicAdd(&out[(size_t)t * NHID + c], wgt * a);
      }
    }
  }
}

__global__ void k_bias_act(float* __restrict__ h, const float* __restrict__ b, int relu) {
  long long i = (long long)blockIdx.x * blockDim.x + threadIdx.x;
  if (i >= (long long)N_NODES * NHID) return;
  float v = h[i] + b[(int)(i & (NHID - 1))];
  if (relu) v = fmaxf(v, 0.0f);
  h[i] = v;
}

// xfin = emb + h(N x 64) @ Wd(64 x 128) + bd ; one wave per 16-row tile
__global__ void __launch_bounds__(32)
k_decode(const float* __restrict__ h, const float* __restrict__ Wd,
         const float* __restrict__ bd, const float* __restrict__ emb,
         float* __restrict__ xfin) {
  int tile  = blockIdx.x;
  int lane  = threadIdx.x;
  int m     = lane & 15;
  int khalf = (lane >> 4) * 2;
  v8f z = {};
  v8f acc[8] = {z, z, z, z, z, z, z, z};
  const float* arow = h + (size_t)(tile * 16 + m) * NHID;
  for (int k = 0; k < NHID; k += 4) {
    int kk = k + khalf;
    v2f a; a.x = arow[kk]; a.y = arow[kk + 1];
    const float* brow0 = Wd + (size_t)kk * NEMB;
    const float* brow1 = Wd + (size_t)(kk + 1) * NEMB;
#pragma unroll
    for (int ob = 0; ob < 8; ++ob) {
      v2f b; b.x = brow0[ob * 16 + m]; b.y = brow1[ob * 16 + m];
      acc[ob] = __builtin_amdgcn_wmma_f32_16x16x4_f32(false, a, false, b, (short)0, acc[ob], false, false);
    }
  }
  int rowSel = (lane < 16) ? 0 : 8;
#pragma unroll
  for (int ob = 0; ob < 8; ++ob)
#pragma unroll
    for (int j = 0; j < 8; ++j) {
      size_t row = (size_t)(tile * 16 + rowSel + j) * NEMB;
      int col = ob * 16 + m;
      xfin[row + col] = emb[row + col] + bd[col] + acc[ob][j];
    }
}

// DistMult score: out[t] = sum_d xfin[s,d] * rel[p,d] * xfin[o,d]
__global__ void k_score(const float* __restrict__ xfin, const float* __restrict__ relEmb,
                        const int* __restrict__ trip, float* __restrict__ out) {
  int t = blockIdx.x * blockDim.x + threadIdx.x;
  if (t >= NTRIP) return;
  int s = trip[3*t], p = trip[3*t+1], o = trip[3*t+2];
  const float4* xs = (const float4*)(xfin   + (size_t)s * NEMB);
  const float4* xp = (const float4*)(relEmb + (size_t)p * NEMB);
  const float4* xo = (const float4*)(xfin   + (size_t)o * NEMB);
  float acc = 0.0f;
#pragma unroll 4
  for (int i = 0; i < NEMB / 4; ++i) {
    float4 a = xs[i], b = xp[i], c = xo[i];
    acc += a.x*b.x*c.x + a.y*b.y*c.y + a.z*b.z*c.z + a.w*b.w*c.w;
  }
  out[t] = acc;
}

extern "C" void kernel_launch(void* const* d_in, const int* in_sizes, int n_in,
                              void* d_out, int out_size, void* d_ws, size_t ws_size,
                              hipStream_t stream) {
  (void)in_sizes; (void)n_in; (void)out_size; (void)ws_size;
  const int*   graph = (const int*)  d_in[0];
  const int*   trips = (const int*)  d_in[1];
  const float* emb   = (const float*)d_in[2];
  const float* rels  = (const float*)d_in[3];
  const float* We    = (const float*)d_in[4];
  const float* be    = (const float*)d_in[5];
  const float* W1    = (const float*)d_in[6];
  const float* b1    = (const float*)d_in[7];
  const float* W2    = (const float*)d_in[8];
  const float* b2    = (const float*)d_in[9];
  const float* Wd    = (const float*)d_in[10];
  const float* bd    = (const float*)d_in[11];
  float* out = (float*)d_out;

  // workspace layout (~86 MB)
  char* ws = (char*)d_ws;
  size_t off = 0;
  float* x0   = (float*)(ws + off); off += (size_t)N_NODES * NHID * 4;
  float* h1   = (float*)(ws + off); off += (size_t)N_NODES * NHID * 4;
  float* h2   = (float*)(ws + off); off += (size_t)N_NODES * NHID * 4;
  float* xfin = (float*)(ws + off); off += (size_t)N_NODES * NEMB * 4;
  float* deg1 = (float*)(ws + off); off += (size_t)N_NODES * 4;
  float* deg2 = (float*)(ws + off); off += (size_t)RTOT * N_NODES * 4;
  int* relCount  = (int*)(ws + off); off += (size_t)RTOT * 4;
  int* relOffset = (int*)(ws + off); off += (size_t)(RTOT + 1) * 4;
  int* relCursor = (int*)(ws + off); off += (size_t)RTOT * 4;
  int* perm      = (int*)(ws + off); off += (size_t)EAUG * 4;

  // zero h1, h2, xfin, deg1, deg2, relCount (contiguous region)
  long long zn = (long long)N_NODES * NHID * 2 + (long long)N_NODES * NEMB
               + N_NODES + (long long)RTOT * N_NODES + RTOT;
  k_zero<<<(unsigned)((zn + 255) / 256), 256, 0, stream>>>(h1, zn);

  unsigned egrid = (EAUG + 255) / 256;
  k_degree<<<egrid, 256, 0, stream>>>(graph, deg1, deg2, relCount);
  k_scan<<<1, 1, 0, stream>>>(relCount, relOffset, relCursor);
  k_bucket<<<egrid, 256, 0, stream>>>(graph, relCursor, perm);

  k_encode<<<N_NODES / 16, 32, 0, stream>>>(emb, We, be, x0);

  unsigned mgrid = (NGROUPS + WPB - 1) / WPB;
  k_message<<<mgrid, 32 * WPB, 0, stream>>>(x0, W1, deg1, deg2, graph, perm, h1, 0);
  k_bias_act<<<(unsigned)(((long long)N_NODES * NHID + 255) / 256), 256, 0, stream>>>(h1, b1, 1);

  k_message<<<mgrid, 32 * WPB, 0, stream>>>(h1, W2, deg1, deg2, graph, perm, h2, 1);
  k_bias_act<<<(unsigned)(((long long)N_NODES * NHID + 255) / 256), 256, 0, stream>>>(h2, b2, 0);

  k_decode<<<N_NODES / 16, 32, 0, stream>>>(h2, Wd, bd, emb, xfin);

  k_score<<<(NTRIP + 255) / 256, 256, 0, stream>>>(xfin, rels, trips, out);
}